// SquareEITNN_3676492005448
// MI455X (gfx1250) — compile-verified
//
#include <hip/hip_runtime.h>
#include <math.h>

// SquareEITNN fused two-layer pipeline for MI455X (gfx1250, wave32).
//
//   hid  = x @ (mask1 ? W1_fixed : W1)^T                      (256 x 9984)
//   outp = hid @ (mask2 ? W2_fixed : atan(Beta)+pi/2)^T       (256 x 9600)
//
// Strategy: memory-bound problem (930 MB weights vs 53 GFLOP). Each workgroup
// owns ONE 16-wide output-column tile and the WHOLE batch (8 waves x 2 WMMA
// accumulator tiles of 16x16), so every weight byte is streamed from HBM once
// with the elementwise transform fused in. fp32 WMMA (16x16x4) keeps exact
// reference precision. atan path is branch-skipped (edges are ~0.02% dense).

typedef __attribute__((ext_vector_type(2))) float v2f;
typedef __attribute__((ext_vector_type(8))) float v8f;

#define BATCH 256
#define INP   768
#define HID   9984
#define OUT   9600
#define HALF_PI 1.57079632679489661923f

// ---------------------------------------------------------------------------
// Layer 1: hid[b][i] = sum_j x[b][j] * (m ? W1f : W1)[i][j]
// grid = HID/16 blocks, 256 threads (8 waves). wave w -> batch rows 32w..32w+31.
// ---------------------------------------------------------------------------
__global__ __launch_bounds__(256)
void eitnn_gemm1(const float* __restrict__ x,
                 const float* __restrict__ W1,
                 const float* __restrict__ W1f,
                 const unsigned char* __restrict__ W1m,
                 float* __restrict__ hid)
{
    const int lane = threadIdx.x & 31;
    const int wave = threadIdx.x >> 5;
    const int i0   = blockIdx.x * 16;        // output-feature tile
    const int n    = lane & 15;              // column within tile / M within A
    const int kh   = (lane >> 4) * 2;        // K sub-pair selector (0 or 2)
    const int b0   = wave * 32;              // two batch tiles per wave

    const size_t wrow = (size_t)(i0 + n) * INP;   // W1 row for this lane's B column
    const float* w1_p  = W1  + wrow;
    const float* w1f_p = W1f + wrow;
    const unsigned char* m_p = W1m + wrow;

    const float* xa0 = x + (size_t)(b0 + n)      * INP;
    const float* xa1 = x + (size_t)(b0 + 16 + n) * INP;

    v8f acc0 = {};
    v8f acc1 = {};

    for (int k0 = 0; k0 < INP; k0 += 4) {
        const int kb = k0 + kh;
        // B pair: W1_dyn[i0+n][kb], [kb+1]  (branchless select, no atan here)
        const unsigned short mm = *(const unsigned short*)(m_p + kb);
        const v2f wdyn = *(const v2f*)(w1_p + kb);
        const v2f wfix = *(const v2f*)(w1f_p + kb);
        v2f b;
        b.x = (mm & 0x0001) ? wfix.x : wdyn.x;
        b.y = (mm & 0x0100) ? wfix.y : wdyn.y;

        const v2f a0 = *(const v2f*)(xa0 + kb);
        const v2f a1 = *(const v2f*)(xa1 + kb);

        acc0 = __builtin_amdgcn_wmma_f32_16x16x4_f32(false, a0, false, b,
                                                     (short)0, acc0, false, false);
        acc1 = __builtin_amdgcn_wmma_f32_16x16x4_f32(false, a1, false, b,
                                                     (short)0, acc1, false, false);
    }

    // D layout: lane<16 -> rows r..r+7 ; lane>=16 -> rows r+8..r+15 ; col = lane&15
    const int col   = i0 + n;
    const int rbase = (lane >> 4) * 8;
#pragma unroll
    for (int r = 0; r < 8; ++r) {
        hid[(size_t)(b0 + rbase + r)      * HID + col] = acc0[r];
        hid[(size_t)(b0 + 16 + rbase + r) * HID + col] = acc1[r];
    }
}

// ---------------------------------------------------------------------------
// Layer 2: outp[b][o] = sum_h hid[b][h] * (m ? W2f : atan(Beta)+pi/2)[o][h]
// grid = OUT/16 blocks. Weights streamed from HBM exactly once; atan branch
// is EXEC-skipped for the ~99.98% of tiles with no graph edges.
// ---------------------------------------------------------------------------
__global__ __launch_bounds__(256)
void eitnn_gemm2(const float* __restrict__ hid,
                 const float* __restrict__ Beta,
                 const float* __restrict__ W2f,
                 const unsigned char* __restrict__ W2m,
                 float* __restrict__ outp)
{
    const int lane = threadIdx.x & 31;
    const int wave = threadIdx.x >> 5;
    const int o0   = blockIdx.x * 16;
    const int n    = lane & 15;
    const int kh   = (lane >> 4) * 2;
    const int b0   = wave * 32;

    const size_t wrow = (size_t)(o0 + n) * HID;
    const float* beta_p = Beta + wrow;
    const float* w2f_p  = W2f  + wrow;
    const unsigned char* m_p = W2m + wrow;

    const float* ha0 = hid + (size_t)(b0 + n)      * HID;
    const float* ha1 = hid + (size_t)(b0 + 16 + n) * HID;

    v8f acc0 = {};
    v8f acc1 = {};

    for (int k0 = 0; k0 < HID; k0 += 4) {
        const int kb = k0 + kh;
        const unsigned short mm = *(const unsigned short*)(m_p + kb);
        v2f b = *(const v2f*)(w2f_p + kb);
        // Rare path: mask byte == 0 -> edge entry -> atan(Beta)+pi/2.
        // Divergent branch; whole-wave skip (s_cbranch_execz) almost always.
        if ((mm & 0x0101u) != 0x0101u) {
            if (!(mm & 0x0001u)) b.x = atanf(beta_p[kb])     + HALF_PI;
            if (!(mm & 0x0100u)) b.y = atanf(beta_p[kb + 1]) + HALF_PI;
        }

        const v2f a0 = *(const v2f*)(ha0 + kb);
        const v2f a1 = *(const v2f*)(ha1 + kb);

        acc0 = __builtin_amdgcn_wmma_f32_16x16x4_f32(false, a0, false, b,
                                                     (short)0, acc0, false, false);
        acc1 = __builtin_amdgcn_wmma_f32_16x16x4_f32(false, a1, false, b,
                                                     (short)0, acc1, false, false);
    }

    const int col   = o0 + n;
    const int rbase = (lane >> 4) * 8;
#pragma unroll
    for (int r = 0; r < 8; ++r) {
        outp[(size_t)(b0 + rbase + r)      * OUT + col] = acc0[r];
        outp[(size_t)(b0 + 16 + rbase + r) * OUT + col] = acc1[r];
    }
}

extern "C" void kernel_launch(void* const* d_in, const int* in_sizes, int n_in,
                              void* d_out, int out_size, void* d_ws, size_t ws_size,
                              hipStream_t stream)
{
    (void)in_sizes; (void)n_in; (void)out_size; (void)d_ws; (void)ws_size;

    const float*         x    = (const float*)d_in[0];
    const float*         W1   = (const float*)d_in[1];
    const float*         W1f  = (const float*)d_in[2];
    const unsigned char* W1m  = (const unsigned char*)d_in[3];
    const float*         Beta = (const float*)d_in[4];
    const float*         W2f  = (const float*)d_in[5];
    const unsigned char* W2m  = (const unsigned char*)d_in[6];

    float* outp = (float*)d_out;                       // [BATCH, OUT]
    float* hid  = outp + (size_t)BATCH * OUT;          // [BATCH, HID] (2nd output)

    eitnn_gemm1<<<HID / 16, 256, 0, stream>>>(x, W1, W1f, W1m, hid);
    eitnn_gemm2<<<OUT / 16, 256, 0, stream>>>(hid, Beta, W2f, W2m, outp);
}